// ForwardGenerator_55439437857557
// MI455X (gfx1250) — compile-verified
//
#include <hip/hip_runtime.h>
#include <hip/hip_bf16.h>

// ---------------------------------------------------------------------------
// ForwardGenerator: 60-step autoregressive MLP scan with per-step BatchNorm.
// Persistent cooperative kernel: 256 WGs x 256 threads (8 waves), wave32.
// Each wave owns 16 batch rows for the whole scan; matrix ops use
// V_WMMA_F32_16X16X4_F32 with pre-swizzled B-operand weights in LDS;
// BN uses LDS ds_add_f32 + global atomics + device-wide barrier.
// ---------------------------------------------------------------------------

namespace {
constexpr int kB   = 32768;   // batch
constexpr int kVE  = 60;      // evaluated variables
constexpr int kL   = 8;       // latents per variable
constexpr int kP   = 4;       // parent window
constexpr int kH   = 64;      // hidden
constexpr int kK1  = kL + kP; // 12
constexpr int kNWG = 256;     // workgroups (co-resident; device-wide barrier)
constexpr int kTPB = 256;     // threads per WG (8 wave32)
constexpr int kRowsPerWG   = kB / kNWG;     // 128
constexpr int kHStride     = 66;            // padded LDS row stride (floats)
constexpr float kEps = 1e-5f;
}

typedef float v2f __attribute__((ext_vector_type(2)));
typedef float v8f __attribute__((ext_vector_type(8)));

__device__ __forceinline__ v8f wmma_f32_k4(v2f a, v2f b, v8f c) {
  // D(16x16,f32) = A(16x4,f32) * B(4x16,f32) + C   -> v_wmma_f32_16x16x4_f32
  return __builtin_amdgcn_wmma_f32_16x16x4_f32(false, a, false, b,
                                               (short)0, c, false, false);
}

__device__ __forceinline__ float fast_tanhf(float v) {
#if __has_builtin(__builtin_amdgcn_tanhf)
  return __builtin_amdgcn_tanhf(v);          // v_tanh_f32 (gfx1250 trans op)
#elif __has_builtin(__builtin_amdgcn_tanh_f32)
  return __builtin_amdgcn_tanh_f32(v);
#else
  return tanhf(v);
#endif
}

__device__ __forceinline__ void grid_barrier(unsigned* cnt, unsigned target) {
  __threadfence();          // flush this thread's global stores (device scope)
  __syncthreads();
  if (threadIdx.x == 0) {
    __hip_atomic_fetch_add(cnt, 1u, __ATOMIC_RELEASE, __HIP_MEMORY_SCOPE_AGENT);
    while (__hip_atomic_load(cnt, __ATOMIC_ACQUIRE, __HIP_MEMORY_SCOPE_AGENT) < target) {
      __builtin_amdgcn_s_sleep(2);
    }
  }
  __syncthreads();
}

__global__ void init_ws_kernel(float* stats, int n, unsigned* cnt) {
  int t = blockIdx.x * blockDim.x + threadIdx.x;
  if (t < n) stats[t] = 0.0f;
  if (t == 0) *cnt = 0u;
}

__global__ __launch_bounds__(kTPB, 1)
void fwdgen_kernel(const float* __restrict__ z,   const float* __restrict__ dov,
                   const float* __restrict__ W1,  const float* __restrict__ b1,
                   const float* __restrict__ g1,  const float* __restrict__ be1,
                   const float* __restrict__ W2,  const float* __restrict__ b2,
                   const float* __restrict__ g2,  const float* __restrict__ be2,
                   const float* __restrict__ Wo,  const float* __restrict__ bo,
                   float* __restrict__ x,
                   float* __restrict__ gs1, float* __restrict__ gq1,
                   float* __restrict__ gs2, float* __restrict__ gq2,
                   unsigned* __restrict__ cnt) {
  // Weights staged in WMMA B-operand pair layout:
  //   lds_wXb[(p*kH + n)*2 + {0,1}] = { W[2p][n], W[2p+1][n] }  (p = K-pair)
  __shared__ __align__(16) float lds_w1b[(kK1 / 2) * kH * 2];   // 6x64x2
  __shared__ __align__(16) float lds_w2b[(kH / 2) * kH * 2];    // 32x64x2
  __shared__ float lds_b1[kH], lds_g1v[kH], lds_be1v[kH];
  __shared__ float lds_b2[kH], lds_g2v[kH], lds_be2v[kH];
  __shared__ float lds_wo[kH];
  __shared__ float lds_bo;
  __shared__ float lds_s1[kH], lds_q1[kH], lds_s2[kH], lds_q2[kH];
  __shared__ float lds_mean[kH], lds_rstd[kH];
  __shared__ __align__(16) float lds_h[kTPB / 32][16 * kHStride]; // 16x64 tiles

  const int tid  = threadIdx.x;
  const int lane = tid & 31;
  const int wv   = tid >> 5;
  const int half = lane >> 4;      // 0: lanes 0-15, 1: lanes 16-31
  const int ln   = lane & 15;
  const int rowbase = blockIdx.x * kRowsPerWG + wv * 16;
  const int arow    = rowbase + ln;               // row addressed by this lane

  // Window registers in WMMA-A chunk layout for K=8..11:
  //   win0 = window[arow][2*half], win1 = window[arow][2*half+1]
  v2f winv = *(const v2f*)(dov + arow * kP + 2 * half);
  float win0 = winv.x, win1 = winv.y;
  // x[:,0:4] = do_vals
  *(v2f*)(x + arow * 64 + 2 * half) = winv;

  unsigned bar = 0;

  #pragma unroll 1
  for (int i = 0; i < kVE; ++i) {
    __syncthreads();  // protect LDS reuse from previous step
    if (tid < kH) { lds_s1[tid] = 0.f; lds_q1[tid] = 0.f;
                    lds_s2[tid] = 0.f; lds_q2[tid] = 0.f; }

    // ---- stage per-variable weights into LDS, swizzled to B-layout ----
    {
      const float* w2s = W2 + i * kH * kH;
      const int n  = tid & 63;        // column, coalesced across the wave
      const int pb = tid >> 6;        // 0..3
      #pragma unroll
      for (int j = 0; j < 8; ++j) {
        const int p = pb + 4 * j;     // K-pair 0..31
        v2f pr;
        pr.x = w2s[(2 * p) * kH + n];
        pr.y = w2s[(2 * p + 1) * kH + n];
        *(v2f*)&lds_w2b[(p * kH + n) * 2] = pr;
      }
    }
    if (tid < kH) {
      const float* w1s = W1 + i * kK1 * kH;
      #pragma unroll
      for (int p = 0; p < kK1 / 2; ++p) {
        v2f pr;
        pr.x = w1s[(2 * p) * kH + tid];
        pr.y = w1s[(2 * p + 1) * kH + tid];
        *(v2f*)&lds_w1b[(p * kH + tid) * 2] = pr;
      }
      lds_b1[tid]   = b1 [i * kH + tid];
      lds_g1v[tid]  = g1 [i * kH + tid];
      lds_be1v[tid] = be1[i * kH + tid];
      lds_b2[tid]   = b2 [i * kH + tid];
      lds_g2v[tid]  = g2 [i * kH + tid];
      lds_be2v[tid] = be2[i * kH + tid];
      lds_wo[tid]   = Wo [i * kH + tid];
    }
    if (tid == 0) lds_bo = bo[i];
    __syncthreads();

    // ================= Layer 1: [16x12] @ [12x64] via 12 WMMAs ============
    v8f acc[4];
    #pragma unroll
    for (int t = 0; t < 4; ++t)
      #pragma unroll
      for (int e = 0; e < 8; ++e) acc[t][e] = 0.f;

    const float* zrow = z + (size_t)arow * (kVE * kL) + i * kL;
    #pragma unroll
    for (int c = 0; c < 3; ++c) {
      v2f a;
      if (c < 2) {
        a = *(const v2f*)(zrow + 4 * c + 2 * half);   // K = 0..7 from z_i
      } else {
        a.x = win0; a.y = win1;                       // K = 8..11 from window
      }
      const int p = 2 * c + half;                     // K-pair index
      #pragma unroll
      for (int t = 0; t < 4; ++t) {
        v2f bb = *(const v2f*)&lds_w1b[(p * kH + 16 * t + ln) * 2];
        acc[t] = wmma_f32_k4(a, bb, acc[t]);
      }
    }

    // bias + tanh + per-feature partial stats (wave tile = 16 rows)
    #pragma unroll
    for (int t = 0; t < 4; ++t) {
      const int n = 16 * t + ln;
      const float bias = lds_b1[n];
      float s = 0.f, q = 0.f;
      #pragma unroll
      for (int v = 0; v < 8; ++v) {
        float val = fast_tanhf(acc[t][v] + bias);
        acc[t][v] = val; s += val; q += val * val;
      }
      s += __shfl_xor(s, 16); q += __shfl_xor(q, 16);
      if (half == 0) { atomicAdd(&lds_s1[n], s); atomicAdd(&lds_q1[n], q); }
    }
    __syncthreads();
    if (tid < kH) {
      atomicAdd(&gs1[i * kH + tid], lds_s1[tid]);
      atomicAdd(&gq1[i * kH + tid], lds_q1[tid]);
    }
    grid_barrier(cnt, kNWG * (++bar));
    if (tid < kH) {
      float s = __hip_atomic_load(&gs1[i * kH + tid], __ATOMIC_RELAXED, __HIP_MEMORY_SCOPE_AGENT);
      float q = __hip_atomic_load(&gq1[i * kH + tid], __ATOMIC_RELAXED, __HIP_MEMORY_SCOPE_AGENT);
      float mn = s * (1.f / kB);
      float vr = q * (1.f / kB) - mn * mn;
      lds_mean[tid] = mn;
      lds_rstd[tid] = rsqrtf(vr + kEps);
    }
    __syncthreads();

    // normalize + write row-major tile to LDS (C-layout -> A-layout transit)
    #pragma unroll
    for (int t = 0; t < 4; ++t) {
      const int n = 16 * t + ln;
      const float mn = lds_mean[n], rs = lds_rstd[n];
      const float gg = lds_g1v[n],  sh = lds_be1v[n];
      #pragma unroll
      for (int v = 0; v < 8; ++v) {
        float h1 = gg * (acc[t][v] - mn) * rs + sh;
        lds_h[wv][(v + 8 * half) * kHStride + n] = h1;
      }
    }
    __syncthreads();

    // ================= Layer 2: [16x64] @ [64x64] via 64 WMMAs ============
    #pragma unroll
    for (int t = 0; t < 4; ++t)
      #pragma unroll
      for (int e = 0; e < 8; ++e) acc[t][e] = 0.f;

    #pragma unroll
    for (int c = 0; c < 16; ++c) {
      const int k0 = 4 * c + 2 * half;
      const int p  = 2 * c + half;
      v2f a = *(const v2f*)&lds_h[wv][ln * kHStride + k0];   // ds_load_b64
      #pragma unroll
      for (int t = 0; t < 4; ++t) {
        v2f bb = *(const v2f*)&lds_w2b[(p * kH + 16 * t + ln) * 2];
        acc[t] = wmma_f32_k4(a, bb, acc[t]);
      }
    }

    #pragma unroll
    for (int t = 0; t < 4; ++t) {
      const int n = 16 * t + ln;
      const float bias = lds_b2[n];
      float s = 0.f, q = 0.f;
      #pragma unroll
      for (int v = 0; v < 8; ++v) {
        float val = fast_tanhf(acc[t][v] + bias);
        acc[t][v] = val; s += val; q += val * val;
      }
      s += __shfl_xor(s, 16); q += __shfl_xor(q, 16);
      if (half == 0) { atomicAdd(&lds_s2[n], s); atomicAdd(&lds_q2[n], q); }
    }
    __syncthreads();
    if (tid < kH) {
      atomicAdd(&gs2[i * kH + tid], lds_s2[tid]);
      atomicAdd(&gq2[i * kH + tid], lds_q2[tid]);
    }
    grid_barrier(cnt, kNWG * (++bar));
    if (tid < kH) {
      float s = __hip_atomic_load(&gs2[i * kH + tid], __ATOMIC_RELAXED, __HIP_MEMORY_SCOPE_AGENT);
      float q = __hip_atomic_load(&gq2[i * kH + tid], __ATOMIC_RELAXED, __HIP_MEMORY_SCOPE_AGENT);
      float mn = s * (1.f / kB);
      float vr = q * (1.f / kB) - mn * mn;
      lds_mean[tid] = mn;
      lds_rstd[tid] = rsqrtf(vr + kEps);
    }
    __syncthreads();

    // normalize h2 -> LDS row-major
    #pragma unroll
    for (int t = 0; t < 4; ++t) {
      const int n = 16 * t + ln;
      const float mn = lds_mean[n], rs = lds_rstd[n];
      const float gg = lds_g2v[n],  sh = lds_be2v[n];
      #pragma unroll
      for (int v = 0; v < 8; ++v) {
        float h2 = gg * (acc[t][v] - mn) * rs + sh;
        lds_h[wv][(v + 8 * half) * kHStride + n] = h2;
      }
    }
    __syncthreads();

    // ======= Output column: col[r] = h2[r,:] . wo + bo (split across halves)
    float partial = 0.f;
    {
      const int base = ln * kHStride + 32 * half;
      #pragma unroll
      for (int j = 0; j < 32; ++j)
        partial = fmaf(lds_h[wv][base + j], lds_wo[32 * half + j], partial);
    }
    partial += __shfl_xor(partial, 16);
    const float col = partial + lds_bo;
    if (half == 0) x[arow * 64 + 4 + i] = col;

    // shift autoregressive window (register-resident, one cross-half shuffle)
    float up = __shfl_xor(win0, 16);   // lower half receives w2 from upper half
    win0 = win1;                       // lower: w1 ; upper: w3
    win1 = half ? col : up;            // lower: w2 ; upper: new col
  }
}

extern "C" void kernel_launch(void* const* d_in, const int* in_sizes, int n_in,
                              void* d_out, int out_size, void* d_ws, size_t ws_size,
                              hipStream_t stream) {
  (void)in_sizes; (void)n_in; (void)out_size; (void)ws_size;
  const float* z   = (const float*)d_in[0];
  const float* dov = (const float*)d_in[1];
  const float* W1  = (const float*)d_in[2];
  const float* b1  = (const float*)d_in[3];
  const float* g1  = (const float*)d_in[4];
  const float* be1 = (const float*)d_in[5];
  const float* W2  = (const float*)d_in[6];
  const float* b2  = (const float*)d_in[7];
  const float* g2  = (const float*)d_in[8];
  const float* be2 = (const float*)d_in[9];
  const float* Wo  = (const float*)d_in[10];
  const float* bo  = (const float*)d_in[11];
  float* x = (float*)d_out;

  const int nstat = kVE * kH;               // 3840 per array
  float* ws  = (float*)d_ws;
  float* gs1 = ws;
  float* gq1 = ws + 1 * nstat;
  float* gs2 = ws + 2 * nstat;
  float* gq2 = ws + 3 * nstat;
  unsigned* cnt = (unsigned*)(ws + 4 * nstat);

  init_ws_kernel<<<(4 * nstat + 255) / 256, 256, 0, stream>>>(ws, 4 * nstat, cnt);
  fwdgen_kernel<<<kNWG, kTPB, 0, stream>>>(z, dov, W1, b1, g1, be1,
                                           W2, b2, g2, be2, Wo, bo,
                                           x, gs1, gq1, gs2, gq2, cnt);
}